// DynamicGCN_27204322853265
// MI455X (gfx1250) — compile-verified
//
#include <hip/hip_runtime.h>
#include <hip/hip_bf16.h>

typedef __attribute__((ext_vector_type(2))) float v2f;
typedef __attribute__((ext_vector_type(8))) float v8f;

#define B_ 64
#define P_ 25
#define D_ 128
#define T_ 120
#define L_ 2
#define KNN 4
#define LN_EPS 1e-5f

// LDS layout (float offsets); all offsets even -> 8B-aligned v2f loads
#define OFF_DIST 0          // 640   (25x25 padded)
#define OFF_ADJ  640        // 32*36 = 1152 (padded stride 36, conflict-free)
#define OFF_DSR  1792       // 40    (D^-1/2, zero-padded)
#define OFF_H    1832       // 32*132 = 4224 (h, padded stride 132)
#define OFF_YT   6056       // 128*36 = 4608 (y transposed: [n][p])
#define OFF_Z    10664      // 32*132 = 4224 (adj@y result)
#define SMEM_TOT 14888      // 59552 bytes static LDS

__global__ __launch_bounds__(256)
void DynamicGCN_gfx1250_kernel(const float* __restrict__ x,
                               const float* __restrict__ dist,
                               const float* __restrict__ W,
                               const float* __restrict__ bias,
                               const float* __restrict__ gamma,
                               const float* __restrict__ beta,
                               float* __restrict__ out)
{
    __shared__ __align__(16) float sm[SMEM_TOT];
    const int tid  = threadIdx.x;
    const int lane = tid & 31;
    const int wave = tid >> 5;
    const int bt   = blockIdx.x;          // b*T + t
    const int b    = bt / T_;
    const int t    = bt - b * T_;

    // ---------------- Phase 0: stage inputs into LDS ----------------
    const float* dg = dist + (size_t)bt * (P_ * P_);
    for (int i = tid; i < P_ * P_; i += 256) sm[OFF_DIST + i] = dg[i];
    for (int i = tid; i < 32 * 36; i += 256) sm[OFF_ADJ + i] = 0.0f;

    const float* xg = x + (size_t)b * P_ * D_ * T_ + t;
    for (int i = tid; i < 32 * D_; i += 256) {
        int p = i >> 7, d = i & 127;
        float v = 0.0f;
        if (p < P_) v = xg[(size_t)(p * D_ + d) * T_];
        sm[OFF_H + p * 132 + d] = v;      // rows 25..31 zero (padding safety)
    }
    __syncthreads();

    // ---------------- kNN adjacency (k=4, exclude self) ----------------
    if (tid < P_) {
        const int p = tid;
        int sel[KNN];
        #pragma unroll
        for (int j = 0; j < KNN; ++j) {
            float best = 3.0e38f; int bi = p;
            for (int q = 0; q < P_; ++q) {
                bool skip = (q == p);
                #pragma unroll
                for (int jj = 0; jj < KNN; ++jj)
                    if (jj < j) skip |= (q == sel[jj]);
                float dv = sm[OFF_DIST + p * P_ + q];
                if (!skip && dv < best) { best = dv; bi = q; }
            }
            sel[j] = bi;
            sm[OFF_ADJ + p * 36 + bi] = 1.0f;
        }
        sm[OFF_ADJ + p * 36 + p] = 1.0f;  // self loop
    }
    __syncthreads();

    // degree -> D^-1/2 (zero-padded out to 40 so normalize never reads OOB)
    if (tid < 40) {
        float r = 0.0f;
        if (tid < P_) {
            float s = 0.0f;
            for (int q = 0; q < P_; ++q) s += sm[OFF_ADJ + tid * 36 + q];
            r = rsqrtf(s);                // deg >= 1 always (self loop)
        }
        sm[OFF_DSR + tid] = r;
    }
    __syncthreads();

    for (int i = tid; i < 32 * 36; i += 256) {
        int p = i / 36, q = i - p * 36;
        sm[OFF_ADJ + i] *= sm[OFF_DSR + p] * sm[OFF_DSR + q];
    }

    // WMMA tiling: wave w owns N-tile n0=w*16, both M-tiles (rows 0-15,16-31)
    const int n0    = wave * 16;
    const int mrow  = lane & 15;
    const int khalf = (lane >> 4) * 2;    // lanes 16-31 hold K+2/K+3
    const int roff  = (lane >> 4) * 8;    // D-matrix: upper lanes hold M+8

    for (int l = 0; l < L_; ++l) {
        __syncthreads();
        const float* Wl = W + l * D_ * D_;

        // ---- matmul 1: y = h @ W   (M=32 padded, N=128, K=128) ----
        {
            v8f c0 = {}; v8f c1 = {};
            for (int k = 0; k < D_; k += 4) {
                const int kr = k + khalf;
                v2f a0 = *(const v2f*)&sm[OFF_H + mrow * 132 + kr];
                v2f a1 = *(const v2f*)&sm[OFF_H + (mrow + 16) * 132 + kr];
                v2f bf;                   // B[k][n] = W[k][n]; column pair from L2
                bf.x = Wl[kr * D_ + n0 + mrow];
                bf.y = Wl[(kr + 1) * D_ + n0 + mrow];
                c0 = __builtin_amdgcn_wmma_f32_16x16x4_f32(false, a0, false, bf,
                                                           (short)0, c0, false, false);
                c1 = __builtin_amdgcn_wmma_f32_16x16x4_f32(false, a1, false, bf,
                                                           (short)0, c1, false, false);
            }
            const int col = n0 + mrow;    // write transposed: yT[n][p]
            #pragma unroll
            for (int i = 0; i < 8; ++i) {
                sm[OFF_YT + col * 36 + (i + roff)]        = c0[i];
                sm[OFF_YT + col * 36 + (16 + i + roff)]   = c1[i];
            }
        }
        __syncthreads();

        // ---- matmul 2: z = adj @ y + bias, ReLU  (K=32 padded) ----
        {
            v8f c0 = {}; v8f c1 = {};
            for (int k = 0; k < 32; k += 4) {
                const int kr = k + khalf;
                v2f a0 = *(const v2f*)&sm[OFF_ADJ + mrow * 36 + kr];
                v2f a1 = *(const v2f*)&sm[OFF_ADJ + (mrow + 16) * 36 + kr];
                v2f bf = *(const v2f*)&sm[OFF_YT + (n0 + mrow) * 36 + kr];
                c0 = __builtin_amdgcn_wmma_f32_16x16x4_f32(false, a0, false, bf,
                                                           (short)0, c0, false, false);
                c1 = __builtin_amdgcn_wmma_f32_16x16x4_f32(false, a1, false, bf,
                                                           (short)0, c1, false, false);
            }
            const int col = n0 + mrow;
            const float bv = bias[l * D_ + col];
            #pragma unroll
            for (int i = 0; i < 8; ++i) {
                sm[OFF_Z + (i + roff) * 132 + col]      = fmaxf(c0[i] + bv, 0.0f);
                sm[OFF_Z + (16 + i + roff) * 132 + col] = fmaxf(c1[i] + bv, 0.0f);
            }
        }
        __syncthreads();

        // ---- LayerNorm over D + residual ----
        if (tid < P_) {
            const int p = tid;
            float s = 0.0f, s2 = 0.0f;
            for (int d = 0; d < D_; ++d) {
                float v = sm[OFF_Z + p * 132 + d];
                s += v; s2 += v * v;
            }
            const float mu  = s * (1.0f / D_);
            const float var = s2 * (1.0f / D_) - mu * mu;
            const float inv = rsqrtf(var + LN_EPS);
            for (int d = 0; d < D_; ++d) {
                float v = (sm[OFF_Z + p * 132 + d] - mu) * inv * gamma[l * D_ + d]
                          + beta[l * D_ + d];
                sm[OFF_H + p * 132 + d] += v;
            }
        }
    }
    __syncthreads();

    // ---------------- store h back to [B,P,D,T] ----------------
    float* og = out + (size_t)b * P_ * D_ * T_ + t;
    for (int i = tid; i < P_ * D_; i += 256) {
        int p = i >> 7, d = i & 127;
        og[(size_t)(p * D_ + d) * T_] = sm[OFF_H + p * 132 + d];
    }
}

extern "C" void kernel_launch(void* const* d_in, const int* in_sizes, int n_in,
                              void* d_out, int out_size, void* d_ws, size_t ws_size,
                              hipStream_t stream) {
    const float* x     = (const float*)d_in[0];
    const float* dist  = (const float*)d_in[1];
    const float* W     = (const float*)d_in[2];
    const float* bias  = (const float*)d_in[3];
    const float* gamma = (const float*)d_in[4];
    const float* beta  = (const float*)d_in[5];
    float* out = (float*)d_out;

    dim3 grid(B_ * T_);   // 7680 blocks, one per (b,t)
    dim3 block(256);      // 8 wave32s
    DynamicGCN_gfx1250_kernel<<<grid, block, 0, stream>>>(x, dist, W, bias, gamma, beta, out);
}